// EncoderGATv2Conv_80015240725031
// MI455X (gfx1250) — compile-verified
//
#include <hip/hip_runtime.h>
#include <math.h>

typedef float v2f __attribute__((ext_vector_type(2)));
typedef float v8f __attribute__((ext_vector_type(8)));
typedef int   v2i __attribute__((ext_vector_type(2)));

#define NEG_SLOPE 0.2f
#define EPSV 1e-16f
#define KMAX 128
#define KPAD 4      // +4 floats/row: fragment reads hit 32 distinct LDS banks

static inline int cdiv(int a, int b) { return (a + b - 1) / b; }

// ---------------------------------------------------------------------------
// Async global->LDS staging helpers (CDNA5 GLOBAL_LOAD_ASYNC_TO_LDS_*,
// tracked by ASYNCcnt). Builtin signatures (from compiler diagnostics):
//   b32: (AS1 int*,  AS3 int*,  imm offset, imm cpol)
//   b64: (AS1 v2i*,  AS3 v2i*,  imm offset, imm cpol)
// Fall back to plain LDS stores if not declared.
// ---------------------------------------------------------------------------
#if __has_builtin(__builtin_amdgcn_global_load_async_to_lds_b64) && \
    __has_builtin(__builtin_amdgcn_global_load_async_to_lds_b32)
#define GAT_ASYNC_LDS 1
#endif

__device__ __forceinline__ void stage_b64(const float* g, float* l) {
#ifdef GAT_ASYNC_LDS
  __builtin_amdgcn_global_load_async_to_lds_b64(
      (__attribute__((address_space(1))) v2i*)(v2i*)const_cast<float*>(g),
      (__attribute__((address_space(3))) v2i*)(v2i*)l, 0, 0);
#else
  *(float2*)l = *(const float2*)g;
#endif
}

__device__ __forceinline__ void stage_b32(const float* g, float* l) {
#ifdef GAT_ASYNC_LDS
  __builtin_amdgcn_global_load_async_to_lds_b32(
      (__attribute__((address_space(1))) int*)(int*)const_cast<float*>(g),
      (__attribute__((address_space(3))) int*)(int*)l, 0, 0);
#else
  *l = *g;
#endif
}

__device__ __forceinline__ void stage_wait() {
#ifdef GAT_ASYNC_LDS
#if __has_builtin(__builtin_amdgcn_s_wait_asynccnt)
  __builtin_amdgcn_s_wait_asynccnt(0);
#else
  asm volatile("s_wait_asynccnt 0x0" ::: "memory");
#endif
#endif
}

// ---------------------------------------------------------------------------
// Dense GEMM with bias: Y[N,M] = X[N,K] @ W[K,M] + b[M]
// Block = 256 threads (8 waves) -> 64x32 output tile (4 row-tiles x 2
// col-tiles). X tile and a *transposed* W tile are staged to LDS with async
// loads, then each wave runs the K-loop with V_WMMA_F32_16X16X4_F32 fed
// entirely from LDS (ds_load_b64 fragments, bank-conflict-free via +4 pad).
// K in {128,64}, M in {128,64}; N multiple of 16 (tail block handles a
// partial row set with a wave-uniform guard so EXEC is all-ones at WMMA).
// ---------------------------------------------------------------------------
__global__ void __launch_bounds__(256)
gat_gemm_bias_wmma(const float* __restrict__ X, const float* __restrict__ W,
                   const float* __restrict__ bias, float* __restrict__ Y,
                   int N, int K, int M) {
  __shared__ float Xs[64 * (KMAX + KPAD)];   // [row][k]  (padded)
  __shared__ float Wt[32 * (KMAX + KPAD)];   // [col][k]  (transposed, padded)

  const int KP = K + KPAD;
  const int tid = threadIdx.x;
  const int rowBase = blockIdx.x * 64;
  const int colBase = blockIdx.y * 32;
  const int rowsRem = N - rowBase;           // 16..64, multiple of 16

  // stage X[rowBase:+64, :] -> Xs, two floats per async op, coalesced in k
  {
    const int kh = K >> 1;
    const int nPairs = 64 * kh;
    for (int idx = tid; idx < nPairs; idx += 256) {
      int row = idx / kh;
      int kk = (idx - row * kh) << 1;
      if (row < rowsRem)
        stage_b64(X + (size_t)(rowBase + row) * K + kk, &Xs[row * KP + kk]);
    }
  }
  // stage W[:, colBase:+32] -> Wt transposed (scatter to LDS is free)
  {
    const int nElems = K * 32;
    for (int idx = tid; idx < nElems; idx += 256) {
      int k = idx >> 5;
      int c = idx & 31;
      stage_b32(W + (size_t)k * M + colBase + c, &Wt[c * KP + k]);
    }
  }
  stage_wait();
  __syncthreads();

  const int wave = tid >> 5;
  const int lane = tid & 31;
  const int rt = wave >> 1;                  // row-tile 0..3
  const int ct = wave & 1;                   // col-tile 0..1
  if (rt * 16 >= rowsRem) return;            // wave-uniform (no barrier after)

  const int l16 = lane & 15;
  const int khalf = (lane >> 4) << 1;        // lanes 0-15: K=k,k+1 ; 16-31: k+2,k+3
  const float* xa = &Xs[(rt * 16 + l16) * KP + khalf];
  const float* wb = &Wt[(ct * 16 + l16) * KP + khalf];

  v8f acc = {};
  #pragma unroll 4
  for (int kb = 0; kb < K; kb += 4) {
    const float2 av = *(const float2*)(xa + kb);
    const float2 bv = *(const float2*)(wb + kb);
    v2f a; a.x = av.x; a.y = av.y;
    v2f b; b.x = bv.x; b.y = bv.y;
    acc = __builtin_amdgcn_wmma_f32_16x16x4_f32(
        /*neg_a=*/false, a, /*neg_b=*/false, b,
        /*c_mod=*/(short)0, acc, /*reuse_a=*/false, /*reuse_b=*/false);
  }

  // D layout: lanes 0-15 -> rows v, lanes 16-31 -> rows v+8 ; col = lane&15
  const int bcol = colBase + ct * 16 + l16;
  const int rbase = rowBase + rt * 16 + ((lane >> 4) << 3);
  const float bv = bias[bcol];
  #pragma unroll
  for (int v = 0; v < 8; ++v)
    Y[(size_t)(rbase + v) * M + bcol] = acc[v] + bv;
}

// ---------------------------------------------------------------------------
// Buffer init: emax = -inf, denom = 0, acc = 0
// ---------------------------------------------------------------------------
__global__ void gat_init(float* __restrict__ emax, float* __restrict__ den,
                         float* __restrict__ acc, int nEH, int nAcc) {
  int i = blockIdx.x * blockDim.x + threadIdx.x;
  if (i < nEH) { emax[i] = -__builtin_inff(); den[i] = 0.0f; }
  if (i < nAcc) acc[i] = 0.0f;
}

__device__ inline void atomicMaxF32(float* addr, float val) {
  unsigned int* ua = (unsigned int*)addr;
  unsigned int cur = *ua;
  while (__uint_as_float(cur) < val) {
    unsigned int prev = atomicCAS(ua, cur, __float_as_uint(val));
    if (prev == cur) break;
    cur = prev;
  }
}

// ---------------------------------------------------------------------------
// Per (edge, head): e = att_h . leaky_relu(xl[src]+xr[dst]) ; atomicMax emax
// ---------------------------------------------------------------------------
__global__ void gat_edge_score(const int* __restrict__ src, const int* __restrict__ dst,
                               const float* __restrict__ XL, const float* __restrict__ XR,
                               const float* __restrict__ att, float* __restrict__ esc,
                               float* __restrict__ emax, int E) {
  int idx = blockIdx.x * blockDim.x + threadIdx.x;
  if (idx >= 2 * E) return;
  int e = idx >> 1, h = idx & 1;
  int s = src[e], d = dst[e];
  const float4* l4 = (const float4*)(XL + (size_t)s * 128 + h * 64);
  const float4* r4 = (const float4*)(XR + (size_t)d * 128 + h * 64);
  const float4* a4 = (const float4*)(att + h * 64);
  float acc = 0.0f;
  #pragma unroll
  for (int c = 0; c < 16; ++c) {
    float4 l = l4[c], r = r4[c], a = a4[c];
    float m;
    m = l.x + r.x; m = m > 0.0f ? m : NEG_SLOPE * m; acc += m * a.x;
    m = l.y + r.y; m = m > 0.0f ? m : NEG_SLOPE * m; acc += m * a.y;
    m = l.z + r.z; m = m > 0.0f ? m : NEG_SLOPE * m; acc += m * a.z;
    m = l.w + r.w; m = m > 0.0f ? m : NEG_SLOPE * m; acc += m * a.w;
  }
  esc[idx] = acc;
  atomicMaxF32(&emax[d * 2 + h], acc);
}

// ---------------------------------------------------------------------------
// Per (edge, head): ee = exp(e - emax[dst]) ; atomicAdd denom
// ---------------------------------------------------------------------------
__global__ void gat_edge_expsum(const int* __restrict__ dst, float* __restrict__ esc,
                                const float* __restrict__ emax, float* __restrict__ den,
                                int E) {
  int idx = blockIdx.x * blockDim.x + threadIdx.x;
  if (idx >= 2 * E) return;
  int e = idx >> 1, h = idx & 1;
  int d = dst[e];
  float ee = __expf(esc[idx] - emax[d * 2 + h]);
  esc[idx] = ee;
  atomicAdd(&den[d * 2 + h], ee);
}

// ---------------------------------------------------------------------------
// One wave per edge; lane owns 4 channels of the 128-wide message:
// acc[dst] += (ee/denom[dst]) * xl[src]
// ---------------------------------------------------------------------------
__global__ void gat_edge_scatter(const int* __restrict__ src, const int* __restrict__ dst,
                                 const float* __restrict__ XL, const float* __restrict__ esc,
                                 const float* __restrict__ den, float* __restrict__ acc,
                                 int E) {
  int gid = blockIdx.x * blockDim.x + threadIdx.x;
  int e = gid >> 5;                   // wave-uniform
  if (e >= E) return;
  int lane = gid & 31;
  int s = src[e], d = dst[e];
  int h = lane >> 4;                  // channels [0,64) head 0, [64,128) head 1
  float alpha = esc[e * 2 + h] / (den[d * 2 + h] + EPSV);
  float4 v = ((const float4*)(XL + (size_t)s * 128))[lane];
  float* out = acc + (size_t)d * 128 + lane * 4;
  atomicAdd(out + 0, alpha * v.x);
  atomicAdd(out + 1, alpha * v.y);
  atomicAdd(out + 2, alpha * v.z);
  atomicAdd(out + 3, alpha * v.w);
}

// ---------------------------------------------------------------------------
// out[n,c] = relu( mean_h acc[n,h,c] + bo[c] + skip[n,c] )
// ---------------------------------------------------------------------------
__global__ void gat_node_out(const float* __restrict__ acc, const float* __restrict__ S,
                             const float* __restrict__ bo, float* __restrict__ out,
                             int N) {
  int i = blockIdx.x * blockDim.x + threadIdx.x;
  if (i >= N * 64) return;
  int n = i >> 6, c = i & 63;
  float v = 0.5f * (acc[(size_t)n * 128 + c] + acc[(size_t)n * 128 + 64 + c])
            + bo[c] + S[i];
  out[i] = v > 0.0f ? v : 0.0f;
}

// ---------------------------------------------------------------------------
extern "C" void kernel_launch(void* const* d_in, const int* in_sizes, int n_in,
                              void* d_out, int out_size, void* d_ws, size_t ws_size,
                              hipStream_t stream) {
  const int N = in_sizes[0] / 128;        // 50000
  const int E = in_sizes[1] / 2;          // 800000

  const float* x   = (const float*)d_in[0];
  const int*   ei  = (const int*)d_in[1];
  const int*   src = ei;
  const int*   dst = ei + E;

  // workspace layout (floats) — total ~27.5M floats (~110 MB); fits in L2.
  float* ws   = (float*)d_ws;
  float* XL   = ws;                         // N*128
  float* XR   = XL  + (size_t)N * 128;      // N*128
  float* S    = XR  + (size_t)N * 128;      // N*64
  float* ESC  = S   + (size_t)N * 64;       // E*2
  float* EMAX = ESC + (size_t)E * 2;        // N*2
  float* DEN  = EMAX + (size_t)N * 2;       // N*2
  float* ACC  = DEN + (size_t)N * 2;        // N*128
  float* HBUF = ACC + (size_t)N * 128;      // N*64

  const float* hin = x;
  int din = 128;

  for (int l = 0; l < 3; ++l) {
    const float* Wl  = (const float*)d_in[2 + l * 8 + 0];
    const float* bl  = (const float*)d_in[2 + l * 8 + 1];
    const float* Wr  = (const float*)d_in[2 + l * 8 + 2];
    const float* br  = (const float*)d_in[2 + l * 8 + 3];
    const float* att = (const float*)d_in[2 + l * 8 + 4];
    const float* bo  = (const float*)d_in[2 + l * 8 + 5];
    const float* Wsk = (const float*)d_in[2 + l * 8 + 6];
    const float* bs  = (const float*)d_in[2 + l * 8 + 7];

    // dense GEMMs (async-LDS staged WMMA): XL = h@Wl+bl, XR = h@Wr+br, S = h@Ws+bs
    {
      dim3 g128(cdiv(N, 64), 128 / 32);
      dim3 g64(cdiv(N, 64), 64 / 32);
      gat_gemm_bias_wmma<<<g128, 256, 0, stream>>>(hin, Wl, bl, XL, N, din, 128);
      gat_gemm_bias_wmma<<<g128, 256, 0, stream>>>(hin, Wr, br, XR, N, din, 128);
      gat_gemm_bias_wmma<<<g64, 256, 0, stream>>>(hin, Wsk, bs, S, N, din, 64);
    }

    gat_init<<<cdiv(N * 128, 256), 256, 0, stream>>>(EMAX, DEN, ACC, N * 2, N * 128);

    gat_edge_score<<<cdiv(2 * E, 256), 256, 0, stream>>>(src, dst, XL, XR, att, ESC, EMAX, E);
    gat_edge_expsum<<<cdiv(2 * E, 256), 256, 0, stream>>>(dst, ESC, EMAX, DEN, E);
    gat_edge_scatter<<<cdiv(E * 32, 256), 256, 0, stream>>>(src, dst, XL, ESC, DEN, ACC, E);

    float* out = (l == 2) ? (float*)d_out : HBUF;
    gat_node_out<<<cdiv(N * 64, 256), 256, 0, stream>>>(ACC, S, bo, out, N);

    hin = HBUF;
    din = 64;
  }
}